// TwikerModel_35175782154709
// MI455X (gfx1250) — compile-verified
//
#include <hip/hip_runtime.h>

typedef __attribute__((ext_vector_type(2))) float v2f;
typedef __attribute__((ext_vector_type(8))) float v8f;

namespace {
constexpr int NB = 4;     // batch
constexpr int NH = 16;    // heads per (key|value) group
constexpr int NG = 32;    // 2*H groups
constexpr int NT = 2048;  // tokens
constexpr int NF = 64;    // features
constexpr int KW = 5;     // conv taps
constexpr int NM = 3;     // boundary variants (P+1)
constexpr int NTILE = NB * NG * (NT / 16);  // 16384 waves; each does full F=64
}

struct F3 { float x, y, z; };  // 12-byte packed store of the 3 m-variants

__device__ __forceinline__ float sel5(const float* wv, int d) {
  // wv indexed only by constants -> stays in VGPRs; d is lane-varying.
  // Compares on d are CSE'd across the three variants by the compiler.
  float t = (d == 0) ? wv[0] : 0.f;
  t = (d == 1) ? wv[1] : t;
  t = (d == 2) ? wv[2] : t;
  t = (d == 3) ? wv[3] : t;
  t = (d == 4) ? wv[4] : t;
  return t;
}

__global__ __launch_bounds__(256)
void twiker_wmma_kernel(const int* __restrict__ ids,
                        const float* __restrict__ keyp,
                        const float* __restrict__ valp,
                        const float* __restrict__ emb,
                        float* __restrict__ out) {
  const int lane = threadIdx.x & 31;
  int tile = blockIdx.x * 8 + (threadIdx.x >> 5);
  const int nblk = tile & 127; tile >>= 7;  // token block 0..127
  const int g    = tile & 31;  tile >>= 5;  // group 0..31
  const int b    = tile;                    // batch 0..3

  const int n0 = nblk * 16;
  const int r  = lane & 15;  // matrix row (A) / column (B,D)
  const int h  = lane >> 4;  // lane-half selects K/M offsets per ISA layout

  // ---- per-row-token softmax numerators (row token = n0 + r) ----
  float ex[KW];
  if (g < NH) {
    // key group: logits [0,0,10,0,0]; the boundary mask multiplies logits,
    // so all three variants share these weights.
    ex[0] = 1.f; ex[1] = 1.f; ex[2] = __expf(10.f); ex[3] = 1.f; ex[4] = 1.f;
  } else {
    const int id = ids[b * NT + n0 + r];
    const float* e = emb + (long)id * (2 * KW) + KW;  // logits = emb[id, 5:10]
#pragma unroll
    for (int k = 0; k < KW; ++k) ex[k] = __expf(e[k]);
  }
  float w[NM][KW];
  {
    const float i0 = 1.f / (ex[0] + ex[1] + ex[2] + ex[3] + ex[4]);
#pragma unroll
    for (int k = 0; k < KW; ++k) w[0][k] = ex[k] * i0;
    const float i1 = 1.f / (2.f + ex[1] + ex[2] + ex[3]);  // masked taps -> e^0
    w[1][0] = i1; w[1][1] = ex[1] * i1; w[1][2] = ex[2] * i1;
    w[1][3] = ex[3] * i1; w[1][4] = i1;
    const float i2 = 1.f / (4.f + ex[2]);
    w[2][0] = i2; w[2][1] = i2; w[2][2] = ex[2] * i2; w[2][3] = i2; w[2][4] = i2;
  }

  // ---- build the three banded A matrices ONCE (reused for all 4 f-blocks).
  // A[M=r][K = p+2h] within chunk cc covers global band column j = 4cc+p+2h;
  // nonzero iff 0 <= j - r < 5 with value w[m][j-r].
  v2f A[NM][KW];
#pragma unroll
  for (int cc = 0; cc < 5; ++cc) {
#pragma unroll
    for (int p = 0; p < 2; ++p) {
      const int d = 4 * cc + p + 2 * h - r;
      A[0][cc][p] = sel5(w[0], d);
      A[1][cc][p] = sel5(w[1], d);
      A[2][cc][p] = sel5(w[2], d);
    }
  }

  // ---- per-slot B row offsets (token halo n0-2 .. n0+17), hoisted out of
  // the feature loop. Only the 2 boundary blocks need clamping/zeroing.
  const float* kv = (g < NH)
      ? keyp + ((long)(b * NH + g) * NT) * NF
      : valp + ((long)(b * NH + (g - NH)) * NT) * NF;
  const bool edge = (nblk == 0) || (nblk == 127);  // wave-uniform

  int rowoff[KW][2];
  bool ok[KW][2];
#pragma unroll
  for (int cc = 0; cc < 5; ++cc) {
#pragma unroll
    for (int p = 0; p < 2; ++p) {
      const int tok = n0 - 2 + 4 * cc + p + 2 * h;
      const int tc  = tok < 0 ? 0 : (tok > NT - 1 ? NT - 1 : tok);
      rowoff[cc][p] = tc * NF;
      ok[cc][p] = (tok == tc);
    }
  }

  float* ob = out + ((long)(b * NG + g) * NT) * NF * NM;

#pragma unroll
  for (int fb = 0; fb < 4; ++fb) {
    const int f0 = fb * 16;

    // B[K=p+2h][N=r] = kv[token, f0+r]
    v2f Bop[5];
    if (edge) {
#pragma unroll
      for (int cc = 0; cc < 5; ++cc)
#pragma unroll
        for (int p = 0; p < 2; ++p) {
          const float vl = kv[rowoff[cc][p] + f0 + r];
          Bop[cc][p] = ok[cc][p] ? vl : 0.f;  // zero-pad halo
        }
    } else {
#pragma unroll
      for (int cc = 0; cc < 5; ++cc)
#pragma unroll
        for (int p = 0; p < 2; ++p)
          Bop[cc][p] = kv[rowoff[cc][p] + f0 + r];
    }

    // D[m] = A[m] (16x20 banded) x B (20x16): 5 chained 16x16x4 WMMAs each.
    v8f acc[NM];
#pragma unroll
    for (int m = 0; m < NM; ++m) {
      v8f c = {};
#pragma unroll
      for (int cc = 0; cc < 5; ++cc) {
        c = __builtin_amdgcn_wmma_f32_16x16x4_f32(
            /*neg_a=*/false, A[m][cc], /*neg_b=*/false, Bop[cc],
            /*c_mod=*/(short)0, c, /*reuse_a=*/false, /*reuse_b=*/false);
      }
      acc[m] = c;
    }

    // D vgpr v -> (token n0+v+8h, feature f0+r); the 3 m-variants are the
    // innermost output dim -> one contiguous 12B packet per (token,feature).
#pragma unroll
    for (int v = 0; v < 8; ++v) {
      const int tokRow = n0 + v + 8 * h;
      F3 pkt;
      pkt.x = acc[0][v]; pkt.y = acc[1][v]; pkt.z = acc[2][v];
      *reinterpret_cast<F3*>(ob + ((long)tokRow * NF + (f0 + r)) * NM) = pkt;
    }
  }
}

extern "C" void kernel_launch(void* const* d_in, const int* in_sizes, int n_in,
                              void* d_out, int out_size, void* d_ws, size_t ws_size,
                              hipStream_t stream) {
  (void)in_sizes; (void)n_in; (void)out_size; (void)d_ws; (void)ws_size;
  const int*   ids = (const int*)d_in[0];
  const float* key = (const float*)d_in[1];
  const float* val = (const float*)d_in[2];
  const float* emb = (const float*)d_in[3];
  float* out = (float*)d_out;
  // 16384 waves (one per b,g,n-block covering all of F), 8 waves per block.
  twiker_wmma_kernel<<<NTILE / 8, 256, 0, stream>>>(ids, key, val, emb, out);
}